// LocallyConnected1D_47124381172270
// MI455X (gfx1250) — compile-verified
//
#include <hip/hip_runtime.h>

// LocallyConnected1D: out[b,l,f] = sum_{k,c} x[b,l+k,c] * W[l,k,c,f] + bias[l,f]
// B=8, L=4096, C_IN=64, K=3, F=64, L_OUT=4094. All fp32.
//
// HBM-bound: kernel weights are 201 MB read exactly once (AI ~7.7 FLOP/B),
// floor ~9us at 23.3 TB/s. x (8.4 MB) is L2-resident. Strategy:
//   - one wave per output position l: 16x64 tile = A(16x192 patch, rows 8..15 pad)
//     x B(192x64 = W_l) via V_WMMA_F32_16X16X4_F32, 48 fully-unrolled k-steps.
//   - weights loaded non-temporally (read-once stream, don't evict x from L2).
//   - the 192-float patch is contiguous (x row stride == C_IN), so A loads are
//     one base + immediate offsets; weight offsets are immediates too -> the
//     unrolled body is pure vmem+wmma with almost no address VALU.

constexpr int Bsz   = 8;
constexpr int Lx    = 4096;
constexpr int C_IN  = 64;
constexpr int KTAPS = 3;
constexpr int Fdim  = 64;
constexpr int L_OUT = (Lx - KTAPS) + 1;   // 4094
constexpr int KC    = KTAPS * C_IN;       // 192

typedef __attribute__((ext_vector_type(2))) float v2f;
typedef __attribute__((ext_vector_type(8))) float v8f;

__global__ __launch_bounds__(256) void
LocallyConnected1D_wmma_f32_kernel(const float* __restrict__ x,
                                   const float* __restrict__ w,
                                   const float* __restrict__ bias,
                                   float* __restrict__ out)
{
    const int lane = threadIdx.x & 31;
    const int wave = threadIdx.x >> 5;
    const int l    = blockIdx.x * 8 + wave;
    if (l >= L_OUT) return;                 // wave-uniform exit

    const int  m  = lane & 15;              // A row index (M) / C column index (N)
    const bool hi = lane >= 16;             // lane half: K pair {2,3} / rows M+8
    const int  b  = m & 7;                  // clamp batch for padded rows 8..15

    // Per-lane base pointers; everything else is a compile-time immediate.
    // A: patch x[b, l .. l+2, :] is 192 contiguous floats at xb.
    const float* xa = x + ((size_t)b * Lx + l) * C_IN + (hi ? 2 : 0);
    // B: element (kidx+kk0)*F + 16n + m ; kk0 = hi ? 2 : 0 folded into base.
    const float* wb = w + (size_t)l * KC * Fdim + (hi ? 2 * Fdim : 0) + m;

    // Seed accumulators with bias[l, 16n + m] (same value down each column)
    v8f acc0, acc1, acc2, acc3;
    {
        const float* bl = bias + (size_t)l * Fdim;
        const float bv0 = bl[ 0 + m];
        const float bv1 = bl[16 + m];
        const float bv2 = bl[32 + m];
        const float bv3 = bl[48 + m];
        #pragma unroll
        for (int r = 0; r < 8; ++r) {
            acc0[r] = bv0; acc1[r] = bv1; acc2[r] = bv2; acc3[r] = bv3;
        }
    }

    #pragma unroll
    for (int s = 0; s < KC / 4; ++s) {
        const int kidx = 4 * s;                 // compile-time constant (full unroll)

        // A fragment: lanes hold x-patch[m][kidx + {0,1} or {2,3}] (b64, imm offset)
        const v2f a = *(const v2f*)(xa + kidx);

        // B fragments: rows kidx+kk striped across lane halves, cols 16n+m.
        // Non-temporal: the 201 MB weight stream is read-once.
        const float* wk = wb + (size_t)kidx * Fdim;
        v2f b0, b1, b2, b3;
        b0.x = __builtin_nontemporal_load(wk + 0);
        b0.y = __builtin_nontemporal_load(wk + Fdim + 0);
        b1.x = __builtin_nontemporal_load(wk + 16);
        b1.y = __builtin_nontemporal_load(wk + Fdim + 16);
        b2.x = __builtin_nontemporal_load(wk + 32);
        b2.y = __builtin_nontemporal_load(wk + Fdim + 32);
        b3.x = __builtin_nontemporal_load(wk + 48);
        b3.y = __builtin_nontemporal_load(wk + Fdim + 48);

        // 8 args: (neg_a, A, neg_b, B, c_mod, C, reuse_a, reuse_b)
        acc0 = __builtin_amdgcn_wmma_f32_16x16x4_f32(false, a, false, b0, (short)0, acc0, false, false);
        acc1 = __builtin_amdgcn_wmma_f32_16x16x4_f32(false, a, false, b1, (short)0, acc1, false, false);
        acc2 = __builtin_amdgcn_wmma_f32_16x16x4_f32(false, a, false, b2, (short)0, acc2, false, false);
        acc3 = __builtin_amdgcn_wmma_f32_16x16x4_f32(false, a, false, b3, (short)0, acc3, false, false);
    }

    // C/D layout: VGPR r, lanes 0..15 -> M=r (valid batches 0..7), N=lane.
    // Rows 8..15 (lanes 16..31) are padding - not stored.
    if (!hi) {
        #pragma unroll
        for (int r = 0; r < 8; ++r) {
            float* orow = out + ((size_t)r * L_OUT + l) * Fdim + m;
            orow[ 0] = acc0[r];
            orow[16] = acc1[r];
            orow[32] = acc2[r];
            orow[48] = acc3[r];
        }
    }
}

extern "C" void kernel_launch(void* const* d_in, const int* in_sizes, int n_in,
                              void* d_out, int out_size, void* d_ws, size_t ws_size,
                              hipStream_t stream) {
    const float* x    = (const float*)d_in[0];  // (8, 4096, 64)
    const float* w    = (const float*)d_in[1];  // (4094, 3, 64, 64)
    const float* bias = (const float*)d_in[2];  // (4094, 64)
    float* out        = (float*)d_out;          // (8, 4094, 64)

    const int waves_per_block = 8;
    const int grid = (L_OUT + waves_per_block - 1) / waves_per_block;  // 512
    LocallyConnected1D_wmma_f32_kernel<<<grid, 256, 0, stream>>>(x, w, bias, out);
}